// LongShortTermTransformer_455266534084
// MI455X (gfx1250) — compile-verified
//
#include <hip/hip_runtime.h>

#define BS_ 2
#define D_  256
#define H_  8
#define FF_ 1024
#define HD_ 32
#define GN_GROUPS_ 32

typedef __attribute__((ext_vector_type(2))) float v2f;
typedef __attribute__((ext_vector_type(8))) float v8f;

// D(16x16,f32) = A(16x4,f32) x B(4x16,f32) + C  -> v_wmma_f32_16x16x4_f32
__device__ __forceinline__ v8f wmma4(v2f a, v2f b, v8f c) {
  return __builtin_amdgcn_wmma_f32_16x16x4_f32(false, a, false, b, (short)0, c,
                                               false, false);
}

// ---------------- LayerNorm (one wave32 per row of 256) ----------------
// y = LN(x (+ x2)) * g + bt ; optional y2 = y + post
template <bool HASX2, bool HASY2>
__global__ void ln_kernel(const float* __restrict__ x, const float* __restrict__ x2,
                          const float* __restrict__ g, const float* __restrict__ bt,
                          float* __restrict__ y, const float* __restrict__ post,
                          float* __restrict__ y2, int rows) {
  int wv = threadIdx.x >> 5, lane = threadIdx.x & 31;
  int row = blockIdx.x * 8 + wv;
  if (row >= rows) return;
  size_t base = (size_t)row * D_;
  float v[8];
  float s = 0.f, ss = 0.f;
#pragma unroll
  for (int i = 0; i < 8; ++i) {
    int c = lane + i * 32;
    float t = x[base + c];
    if (HASX2) t += x2[base + c];
    v[i] = t; s += t; ss += t * t;
  }
#pragma unroll
  for (int m = 16; m >= 1; m >>= 1) {
    s += __shfl_xor(s, m, 32);
    ss += __shfl_xor(ss, m, 32);
  }
  float mean = s * (1.f / D_);
  float var = ss * (1.f / D_) - mean * mean;
  float inv = rsqrtf(var + 1e-5f);
#pragma unroll
  for (int i = 0; i < 8; ++i) {
    int c = lane + i * 32;
    float t = (v[i] - mean) * inv * g[c] + bt[c];
    y[base + c] = t;
    if (HASY2) y2[base + c] = t + post[base + c];
  }
}

// ---------------- GEMM: C = (A (+A2)) @ W + bias (+ accum) ----------------
// A: rows x M, W: M x Kc.  grid.x = rows/16, grid.y covers Kc in 256-wide
// block stripes; block = 128 threads = 4 waves, each wave does 16x64.
template <bool HASA2>
__global__ void gemm_bias_kernel(const float* __restrict__ A, const float* __restrict__ A2,
                                 const float* __restrict__ W, const float* __restrict__ bias,
                                 const float* __restrict__ accum, float* __restrict__ C,
                                 int M, int Kc) {
  int wv = threadIdx.x >> 5, lane = threadIdx.x & 31;
  int hf = lane >> 4, l16 = lane & 15, ka = hf * 2;
  int rowT = blockIdx.x * 16;
  int colBase = (blockIdx.y * 4 + wv) * 64;
  if (colBase >= Kc) return;          // wave-uniform; EXEC stays all-1s below
  int row = rowT + l16;
  v8f acc[4] = {{}, {}, {}, {}};
  for (int k = 0; k < M; k += 4) {
    v2f a;
    size_t ai = (size_t)row * M + k + ka;
    a.x = A[ai];
    a.y = A[ai + 1];
    if (HASA2) { a.x += A2[ai]; a.y += A2[ai + 1]; }
#pragma unroll
    for (int j = 0; j < 4; ++j) {
      int col = colBase + j * 16 + l16;
      v2f b;
      b.x = W[(size_t)(k + ka) * Kc + col];
      b.y = W[(size_t)(k + ka + 1) * Kc + col];
      acc[j] = wmma4(a, b, acc[j]);
    }
  }
#pragma unroll
  for (int j = 0; j < 4; ++j) {
    int col = colBase + j * 16 + l16;
#pragma unroll
    for (int r = 0; r < 8; ++r) {
      int m = rowT + r + 8 * hf;
      float vv = acc[j][r] + bias[col];
      if (accum) vv += accum[(size_t)m * Kc + col];
      C[(size_t)m * Kc + col] = vv;
    }
  }
}

// ---------------- Flash attention (fp32 WMMA, online softmax) ----------------
// Q,K,V,O: (L, BS, D) fp32, head h uses channels [h*32, h*32+32).
// grid = (Lq/16, BS*H), block = 32 (one wave).
__global__ void flash_attn_kernel(const float* __restrict__ Q, const float* __restrict__ K,
                                  const float* __restrict__ V, float* __restrict__ O,
                                  int Lk, float scale) {
  __shared__ float P[16 * 16];
  int lane = threadIdx.x;
  int hf = lane >> 4, l16 = lane & 15, ka = hf * 2;
  int qt = blockIdx.x;
  int bb = blockIdx.y / H_, hh = blockIdx.y % H_;
  const int ld = BS_ * D_;
  int coff = bb * D_ + hh * HD_;

  // preload Q tile (16 x 32) as 8 A-fragments, pre-scaled by 1/sqrt(hd)
  v2f qa[8];
  int qrow = qt * 16 + l16;
  const float* qp = Q + (size_t)qrow * ld + coff;
#pragma unroll
  for (int kc = 0; kc < 8; ++kc) {
    qa[kc].x = qp[kc * 4 + ka] * scale;
    qa[kc].y = qp[kc * 4 + ka + 1] * scale;
  }

  v8f o0 = {}, o1 = {};
  float mrow[8], lrow[8];
#pragma unroll
  for (int r = 0; r < 8; ++r) { mrow[r] = -1e30f; lrow[r] = 0.f; }

  int nkt = Lk / 16;
  for (int kt = 0; kt < nkt; ++kt) {
    // prefetch next key tile's K/V rows (one cacheline each; row stride 2KB)
    // -> global_prefetch_b8; overlaps next tile's L2->WGP fill with softmax VALU
    if (kt + 1 < nkt) {
      const float* knx = K + (size_t)((kt + 1) * 16 + l16) * ld + coff;
      const float* vnx = V + (size_t)((kt + 1) * 16 + l16) * ld + coff;
      __builtin_prefetch(knx, 0, 3);
      __builtin_prefetch(vnx, 0, 3);
    }
    // S tile = Q @ K^T (16x16), K=32 in 8 steps of 4
    v8f s = {};
    const float* kp = K + (size_t)(kt * 16 + l16) * ld + coff;
#pragma unroll
    for (int kc = 0; kc < 8; ++kc) {
      v2f bf;
      bf.x = kp[kc * 4 + ka];
      bf.y = kp[kc * 4 + ka + 1];
      s = wmma4(qa[kc], bf, s);
    }
    // online softmax per row (rows split across 16-lane halves)
    float corr[8], p[8];
#pragma unroll
    for (int r = 0; r < 8; ++r) {
      float vv = s[r];
      float mx = vv;
#pragma unroll
      for (int m = 8; m >= 1; m >>= 1) mx = fmaxf(mx, __shfl_xor(mx, m, 32));
      float mnew = fmaxf(mrow[r], mx);
      float c = __expf(mrow[r] - mnew);
      float pv = __expf(vv - mnew);
      float sum = pv;
#pragma unroll
      for (int m = 8; m >= 1; m >>= 1) sum += __shfl_xor(sum, m, 32);
      lrow[r] = lrow[r] * c + sum;
      mrow[r] = mnew;
      corr[r] = c;
      p[r] = pv;
    }
#pragma unroll
    for (int r = 0; r < 8; ++r) { o0[r] *= corr[r]; o1[r] *= corr[r]; }
    // stage P through LDS (C-layout -> A-layout re-fragmentation)
#pragma unroll
    for (int r = 0; r < 8; ++r) P[(r + 8 * hf) * 16 + l16] = p[r];
    __syncthreads();
    const float* vb = V + coff;
#pragma unroll
    for (int kc2 = 0; kc2 < 4; ++kc2) {
      v2f pa;
      pa.x = P[l16 * 16 + kc2 * 4 + ka];
      pa.y = P[l16 * 16 + kc2 * 4 + ka + 1];
      int keyr = kt * 16 + kc2 * 4 + ka;
      v2f v0, v1;
      v0.x = vb[(size_t)keyr * ld + l16];
      v0.y = vb[(size_t)(keyr + 1) * ld + l16];
      v1.x = vb[(size_t)keyr * ld + 16 + l16];
      v1.y = vb[(size_t)(keyr + 1) * ld + 16 + l16];
      o0 = wmma4(pa, v0, o0);
      o1 = wmma4(pa, v1, o1);
    }
    __syncthreads();
  }
#pragma unroll
  for (int r = 0; r < 8; ++r) {
    float inv = 1.f / lrow[r];
    int m = qt * 16 + r + 8 * hf;
    O[(size_t)m * ld + coff + l16] = o0[r] * inv;
    O[(size_t)m * ld + coff + 16 + l16] = o1[r] * inv;
  }
}

// ---------------- GroupNorm(32) + exact GELU ----------------
// x: (L, BS, FF); one block per (b, group); block = 256.
__global__ void gn_gelu_kernel(const float* __restrict__ x, const float* __restrict__ gw,
                               const float* __restrict__ gb, float* __restrict__ y, int L) {
  __shared__ float red[16];
  int b = blockIdx.x / GN_GROUPS_;
  int g = blockIdx.x % GN_GROUPS_;
  const int cpg = FF_ / GN_GROUPS_;  // 32
  int tot = L * cpg;
  float s = 0.f, ss = 0.f;
  for (int i = threadIdx.x; i < tot; i += blockDim.x) {
    int l = i / cpg, cc = i % cpg;
    float v = x[(size_t)l * (BS_ * FF_) + b * FF_ + g * cpg + cc];
    s += v; ss += v * v;
  }
  int wv = threadIdx.x >> 5, lane = threadIdx.x & 31;
#pragma unroll
  for (int m = 16; m >= 1; m >>= 1) {
    s += __shfl_xor(s, m, 32);
    ss += __shfl_xor(ss, m, 32);
  }
  if (lane == 0) { red[wv] = s; red[8 + wv] = ss; }
  __syncthreads();
  if (threadIdx.x == 0) {
    float S = 0.f, SS = 0.f;
    for (int w = 0; w < 8; ++w) { S += red[w]; SS += red[8 + w]; }
    red[0] = S; red[8] = SS;
  }
  __syncthreads();
  float mean = red[0] / (float)tot;
  float var = red[8] / (float)tot - mean * mean;
  float inv = rsqrtf(var + 1e-5f);
  for (int i = threadIdx.x; i < tot; i += blockDim.x) {
    int l = i / cpg, cc = i % cpg;
    int c = g * cpg + cc;
    size_t idx = (size_t)l * (BS_ * FF_) + b * FF_ + c;
    float v = (x[idx] - mean) * inv * gw[c] + gb[c];
    v = 0.5f * v * (1.f + erff(v * 0.70710678118654752f));
    y[idx] = v;
  }
}

// ---------------- 5x5 depthwise conv, pad 2 ----------------
__global__ void dwconv5_kernel(const float* __restrict__ x, const float* __restrict__ w,
                               float* __restrict__ y, const int* __restrict__ hp,
                               const int* __restrict__ wp, int total) {
  int idx = blockIdx.x * blockDim.x + threadIdx.x;
  if (idx >= total) return;
  int Hh = *hp, Ww = *wp;
  int c = idx % FF_;
  int b = (idx / FF_) % BS_;
  int l = idx / (FF_ * BS_);
  int xx = l % Ww, yy = l / Ww;
  float acc = 0.f;
#pragma unroll
  for (int dy = 0; dy < 5; ++dy) {
    int sy = yy + dy - 2;
    if (sy < 0 || sy >= Hh) continue;
#pragma unroll
    for (int dx = 0; dx < 5; ++dx) {
      int sx = xx + dx - 2;
      if (sx < 0 || sx >= Ww) continue;
      acc += x[(size_t)(sy * Ww + sx) * (BS_ * FF_) + b * FF_ + c] *
             w[c * 25 + dy * 5 + dx];
    }
  }
  y[idx] = acc;
}

extern "C" void kernel_launch(void* const* d_in, const int* in_sizes, int n_in,
                              void* d_out, int out_size, void* d_ws, size_t ws_size,
                              hipStream_t stream) {
  (void)n_in; (void)out_size; (void)ws_size;
  const float* tgt   = (const float*)d_in[0];
  const float* idemb = (const float*)d_in[1];
  const float* spos  = (const float*)d_in[2];
  const float* n1w = (const float*)d_in[3];  const float* n1b = (const float*)d_in[4];
  const float* n2w = (const float*)d_in[5];  const float* n2b = (const float*)d_in[6];
  const float* n3w = (const float*)d_in[7];  const float* n3b = (const float*)d_in[8];
  const float* n4w = (const float*)d_in[9];  const float* n4b = (const float*)d_in[10];
  const float* saqw = (const float*)d_in[11]; const float* saqb = (const float*)d_in[12];
  const float* sakw = (const float*)d_in[13]; const float* sakb = (const float*)d_in[14];
  const float* savw = (const float*)d_in[15]; const float* savb = (const float*)d_in[16];
  const float* sapw = (const float*)d_in[17]; const float* sapb = (const float*)d_in[18];
  const float* ltpw = (const float*)d_in[19]; const float* ltpb = (const float*)d_in[20];
  const float* stpw = (const float*)d_in[21]; const float* stpb = (const float*)d_in[22];
  const float* lqw  = (const float*)d_in[23]; const float* lqb  = (const float*)d_in[24];
  const float* lvw  = (const float*)d_in[25]; const float* lvb  = (const float*)d_in[26];
  const float* ff1w = (const float*)d_in[27]; const float* ff1b = (const float*)d_in[28];
  const float* gnw  = (const float*)d_in[29]; const float* gnb  = (const float*)d_in[30];
  const float* dww  = (const float*)d_in[31];
  const float* ff2w = (const float*)d_in[32]; const float* ff2b = (const float*)d_in[33];
  const float* fnw  = (const float*)d_in[34]; const float* fnb  = (const float*)d_in[35];
  const int* hp = (const int*)d_in[36];
  const int* wp = (const int*)d_in[37];

  const int L = in_sizes[0] / (BS_ * D_);   // 1600
  const int N = L * BS_;                    // 3200 rows
  const size_t DB = (size_t)N * D_;
  const size_t FB = (size_t)N * FF_;
  float* ws = (float*)d_ws;
  float* cur = ws;
  float* b1 = ws + 1 * DB;
  float* b2 = ws + 2 * DB;
  float* b3 = ws + 3 * DB;
  float* b4 = ws + 4 * DB;
  float* b5 = ws + 5 * DB;
  float* b6 = ws + 6 * DB;
  float* b7 = ws + 7 * DB;
  float* f0 = ws + 8 * DB;
  float* f1 = f0 + FB;

  hipMemcpyAsync(cur, tgt, DB * sizeof(float), hipMemcpyDeviceToDevice, stream);

  dim3 lnGrid((N + 7) / 8), lnBlk(256);
  dim3 gmBlk(128);
  dim3 gmGridD(N / 16, 1);                    // Kc = 256 (4 waves * 64)
  dim3 gmGridF(N / 16, FF_ / 256);            // Kc = 1024
  dim3 faGrid(L / 16, BS_ * H_), faBlk(32);
  dim3 gnGrid(BS_ * GN_GROUPS_), gnBlk(256);
  int total = N * FF_;
  dim3 dwGrid((total + 255) / 256), dwBlk(256);
  const float scale = 0.17677669529663687f;   // 1/sqrt(32)

  for (int i = 0; i < 2; ++i) {
    // ---- self attention ----
    // _t = LN(cur, n1) -> b1 ; q = _t + self_pos -> b2
    ln_kernel<false, true><<<lnGrid, lnBlk, 0, stream>>>(cur, nullptr, n1w + i * D_,
                                                         n1b + i * D_, b1, spos, b2, N);
    gemm_bias_kernel<false><<<gmGridD, gmBlk, 0, stream>>>(b2, nullptr, saqw + i * D_ * D_,
                                                           saqb + i * D_, nullptr, b3, D_, D_);
    gemm_bias_kernel<false><<<gmGridD, gmBlk, 0, stream>>>(b2, nullptr, sakw + i * D_ * D_,
                                                           sakb + i * D_, nullptr, b4, D_, D_);
    gemm_bias_kernel<false><<<gmGridD, gmBlk, 0, stream>>>(b1, nullptr, savw + i * D_ * D_,
                                                           savb + i * D_, nullptr, b5, D_, D_);
    flash_attn_kernel<<<faGrid, faBlk, 0, stream>>>(b3, b4, b5, b6, L, scale);
    gemm_bias_kernel<false><<<gmGridD, gmBlk, 0, stream>>>(b6, nullptr, sapw + i * D_ * D_,
                                                           sapb + i * D_, cur, cur, D_, D_);
    // ---- long/short memory attention (first-frame path) ----
    ln_kernel<false, false><<<lnGrid, lnBlk, 0, stream>>>(cur, nullptr, n2w + i * D_,
                                                          n2b + i * D_, b1, nullptr,
                                                          nullptr, N);            // _t2
    gemm_bias_kernel<false><<<gmGridD, gmBlk, 0, stream>>>(b1, nullptr, lqw + i * D_ * D_,
                                                           lqb + i * D_, nullptr, b2,
                                                           D_, D_);               // curr_Q
    gemm_bias_kernel<true><<<gmGridD, gmBlk, 0, stream>>>(b1, idemb, lvw + i * D_ * D_,
                                                          lvb + i * D_, nullptr, b3,
                                                          D_, D_);                // global_V
    flash_attn_kernel<<<faGrid, faBlk, 0, stream>>>(b2, b2, b3, b4, L, scale);
    gemm_bias_kernel<false><<<gmGridD, gmBlk, 0, stream>>>(b4, nullptr, ltpw + i * D_ * D_,
                                                           ltpb + i * D_, cur, cur, D_, D_);
    // k_st = LN(2*curr_Q, n4) ; v_st = LN(global_V + _t2, n4)
    ln_kernel<true, false><<<lnGrid, lnBlk, 0, stream>>>(b2, b2, n4w + i * D_, n4b + i * D_,
                                                         b5, nullptr, nullptr, N);
    ln_kernel<true, false><<<lnGrid, lnBlk, 0, stream>>>(b3, b1, n4w + i * D_, n4b + i * D_,
                                                         b6, nullptr, nullptr, N);
    flash_attn_kernel<<<faGrid, faBlk, 0, stream>>>(b2, b5, b6, b7, L, scale);
    gemm_bias_kernel<false><<<gmGridD, gmBlk, 0, stream>>>(b7, nullptr, stpw + i * D_ * D_,
                                                           stpb + i * D_, cur, cur, D_, D_);
    // ---- FFN: GN + GELU + DWConv ----
    ln_kernel<false, false><<<lnGrid, lnBlk, 0, stream>>>(cur, nullptr, n3w + i * D_,
                                                          n3b + i * D_, b1, nullptr,
                                                          nullptr, N);
    gemm_bias_kernel<false><<<gmGridF, gmBlk, 0, stream>>>(b1, nullptr, ff1w + i * D_ * FF_,
                                                           ff1b + i * FF_, nullptr, f0,
                                                           D_, FF_);
    gn_gelu_kernel<<<gnGrid, gnBlk, 0, stream>>>(f0, gnw + i * FF_, gnb + i * FF_, f1, L);
    dwconv5_kernel<<<dwGrid, dwBlk, 0, stream>>>(f1, dww + i * FF_ * 25, f0, hp, wp, total);
    gemm_bias_kernel<false><<<gmGridD, gmBlk, 0, stream>>>(f0, nullptr, ff2w + i * FF_ * D_,
                                                           ff2b + i * D_, cur, cur, FF_, D_);
  }
  // final decoder norm -> d_out
  ln_kernel<false, false><<<lnGrid, lnBlk, 0, stream>>>(cur, nullptr, fnw, fnb,
                                                        (float*)d_out, nullptr, nullptr, N);
}